// EMDLosspy_16673063043241
// MI455X (gfx1250) — compile-verified
//
#include <hip/hip_runtime.h>

// ---------------------------------------------------------------------------
// Sinkhorn EMD for B=16, N=M=2048, 3-D points, eps=0.02, 50 iterations.
// Cost matrix recomputed on the fly from LDS-resident point sets; the
// -2*x.y rank-4 GEMM per 16x16 tile uses V_WMMA_F32_16X16X4_F32 (packed
// float4 B operand -> one ds_load_b64 per tile).  Branchless grouped
// logsumexp: 4 tiles per running-max merge -> 2.5 trans ops per element
// with fully batched sqrt/exp chains (no exposed trans latency).
// ---------------------------------------------------------------------------

typedef float v2f __attribute__((ext_vector_type(2)));
typedef float v8f __attribute__((ext_vector_type(8)));

#define NPTS   2048
#define NBATCH 16
#define ROWS_PER_BLOCK 128
#define THREADS 256
#define NP_PAD 2064               // 2048 + 16: shifts LDS bank phase
#define GRP 4                     // tiles per logsumexp merge (4 = best schedule)

#define K_B2     72.13475204444817f      // log2(e) / eps  (eps = 0.02)
#define EPS_LN2  0.013862943611198906f   // eps * ln(2)
#define NEG_EPS_LOGN (-0.15249237972318797f) // -eps * ln(2048)

#if defined(__has_builtin)
#  if __has_builtin(__builtin_amdgcn_wmma_f32_16x16x4_f32)
#    define HAVE_WMMA_F32X4 1
#  endif
#endif
#ifndef HAVE_WMMA_F32X4
#  define HAVE_WMMA_F32X4 0
#endif

__device__ __forceinline__ float fast_exp2(float x) { return __builtin_amdgcn_exp2f(x); }
__device__ __forceinline__ float fast_log2(float x) { return __builtin_amdgcn_logf(x); }
__device__ __forceinline__ float fast_sqrt(float x) { return __builtin_amdgcn_sqrtf(x); }

// v_max_num_f32 without LLVM's sNaN-canonicalize preamble.
__device__ __forceinline__ float fmax_raw(float a, float b) {
    float r;
    asm("v_max_num_f32 %0, %1, %2" : "=v"(r) : "v"(a), "v"(b));
    return r;
}

// One Sinkhorn half-step:
//   pot_out[b, n] = eps*log(1/N) - eps * LSE_m( (pot_in[b,m] - C[b,n,m]) / eps )
__global__ __launch_bounds__(THREADS)
void sinkhorn_half(const float* __restrict__ pr,
                   const float* __restrict__ pc,
                   const float* __restrict__ pot_in,
                   float* __restrict__ pot_out)
{
    // Packed B operand per column: [-2y0, -2y1, -2y2, |y|^2]
    __shared__ float4 lbp[NPTS];
    __shared__ float  lg[NP_PAD];
    __shared__ float  lx0[ROWS_PER_BLOCK], lx1[ROWS_PER_BLOCK],
                      lx2[ROWS_PER_BLOCK], lxn[ROWS_PER_BLOCK];

    const int b   = blockIdx.y;
    const int tid = threadIdx.x;
    const int n0b = blockIdx.x * ROWS_PER_BLOCK;

    for (int i = tid; i < NPTS; i += THREADS) {
        const float q0 = pc[(b * NPTS + i) * 3 + 0];
        const float q1 = pc[(b * NPTS + i) * 3 + 1];
        const float q2 = pc[(b * NPTS + i) * 3 + 2];
        lbp[i] = make_float4(-2.0f * q0, -2.0f * q1, -2.0f * q2,
                             q0 * q0 + q1 * q1 + q2 * q2);
        lg[i]  = pot_in[b * NPTS + i] * K_B2;    // prescaled by log2e/eps
    }
    for (int i = tid; i < ROWS_PER_BLOCK; i += THREADS) {
        const float p0 = pr[(b * NPTS + n0b + i) * 3 + 0];
        const float p1 = pr[(b * NPTS + n0b + i) * 3 + 1];
        const float p2 = pr[(b * NPTS + n0b + i) * 3 + 2];
        lx0[i] = p0; lx1[i] = p1; lx2[i] = p2;
        lxn[i] = p0 * p0 + p1 * p1 + p2 * p2;
    }
    __syncthreads();

    const int lane  = tid & 31;
    const int wave  = tid >> 5;
    const int half  = lane >> 4;
    const int l     = lane & 15;
    const int rbase = wave * 16;      // 16 rows per wave

    const float clamp_lo = 1e-12f;

#if HAVE_WMMA_F32X4
    // A fragment (16x4): row M = l; VGPR0 holds K0 (half 0) / K2 (half 1),
    // VGPR1 holds K1 / K3.  A row = [x0, x1, x2, 1].
    v2f afrag;
    afrag.x = half ? lx2[rbase + l] : lx0[rbase + l];
    afrag.y = half ? 1.0f           : lx1[rbase + l];
    // C fragment: |x_row|^2, row = v + 8*half.
    v8f cfrag;
    #pragma unroll
    for (int v = 0; v < 8; ++v) cfrag[v] = lxn[rbase + v + 8 * half];
    // B fragment: one aligned b64 load per tile: {K0,K1} or {K2,K3}.
    const float* bbase = (const float*)lbp + 2 * half;
#else
    float xr0[8], xr1[8], xr2[8], xrn[8];
    #pragma unroll
    for (int v = 0; v < 8; ++v) {
        const int r = rbase + v + 8 * half;
        xr0[v] = lx0[r]; xr1[v] = lx1[r]; xr2[v] = lx2[r]; xrn[v] = lxn[r];
    }
#endif

    // Running base-2 logsumexp accumulators, one per owned row.
    float mx[8], sm[8];
    #pragma unroll
    for (int v = 0; v < 8; ++v) { mx[v] = -3.0e38f; sm[v] = 0.0f; }

    for (int mt = 0; mt < NPTS / 16; mt += GRP) {
        // Phase 1: GRP tiles of WMMA + sqrt + t = (g - d)*log2e/eps.
        float t[GRP][8];
        #pragma unroll
        for (int j = 0; j < GRP; ++j) {
            const int   m  = (mt + j) * 16 + l;
            const float gk = lg[m];
#if HAVE_WMMA_F32X4
            const v2f bfrag = *(const v2f*)(bbase + 4 * m);
            v8f sq = __builtin_amdgcn_wmma_f32_16x16x4_f32(
                false, afrag, false, bfrag, (short)0, cfrag, false, false);
#else
            const float4 yq = lbp[m];
            v8f sq;
            #pragma unroll
            for (int v = 0; v < 8; ++v)
                sq[v] = xrn[v] + yq.w + xr0[v] * yq.x + xr1[v] * yq.y + xr2[v] * yq.z;
#endif
            #pragma unroll
            for (int v = 0; v < 8; ++v) {
                const float dd = fast_sqrt(fmax_raw(sq[v], clamp_lo));
                t[j][v] = __builtin_fmaf(dd, -K_B2, gk);
            }
        }
        // Phase 2: branchless grouped LSE: tree-max over the GRP values,
        // GRP exps vs the group max, one running-accumulator merge (2 exps).
        #pragma unroll
        for (int v = 0; v < 8; ++v) {
            const float tm  = fmax_raw(fmax_raw(t[0][v], t[1][v]),
                                       fmax_raw(t[2][v], t[3][v]));
            const float grp = (fast_exp2(t[0][v] - tm) + fast_exp2(t[1][v] - tm))
                            + (fast_exp2(t[2][v] - tm) + fast_exp2(t[3][v] - tm));
            const float nm  = fmax_raw(mx[v], tm);
            const float eo  = fast_exp2(mx[v] - nm);
            const float en  = fast_exp2(tm - nm);
            sm[v] = __builtin_fmaf(sm[v], eo, grp * en);
            mx[v] = nm;
        }
    }

    // Combine partial LSEs across the 16 lanes of each half.
    #pragma unroll
    for (int v = 0; v < 8; ++v) {
        #pragma unroll
        for (int off = 8; off >= 1; off >>= 1) {
            const float omx = __shfl_xor(mx[v], off, 32);
            const float osm = __shfl_xor(sm[v], off, 32);
            const float nm  = fmax_raw(mx[v], omx);
            sm[v] = sm[v] * fast_exp2(mx[v] - nm) + osm * fast_exp2(omx - nm);
            mx[v] = nm;
        }
    }

    if (l == 0) {
        #pragma unroll
        for (int v = 0; v < 8; ++v) {
            const int n = n0b + rbase + v + 8 * half;
            pot_out[b * NPTS + n] =
                NEG_EPS_LOGN - EPS_LN2 * (mx[v] + fast_log2(sm[v]));
        }
    }
}

// Final loss: out += (1/B) * sum_{n,m} exp((f_n + g_m - C)/eps) * C
__global__ __launch_bounds__(THREADS)
void emd_loss(const float* __restrict__ pr,
              const float* __restrict__ pc,
              const float* __restrict__ fpot,
              const float* __restrict__ gpot,
              float* __restrict__ out)
{
    __shared__ float4 lbp[NPTS];
    __shared__ float  lg[NP_PAD];
    __shared__ float  lx0[ROWS_PER_BLOCK], lx1[ROWS_PER_BLOCK],
                      lx2[ROWS_PER_BLOCK], lxn[ROWS_PER_BLOCK], lf[ROWS_PER_BLOCK];
    __shared__ float  wsum[THREADS / 32];

    const int b   = blockIdx.y;
    const int tid = threadIdx.x;
    const int n0b = blockIdx.x * ROWS_PER_BLOCK;

    for (int i = tid; i < NPTS; i += THREADS) {
        const float q0 = pc[(b * NPTS + i) * 3 + 0];
        const float q1 = pc[(b * NPTS + i) * 3 + 1];
        const float q2 = pc[(b * NPTS + i) * 3 + 2];
        lbp[i] = make_float4(-2.0f * q0, -2.0f * q1, -2.0f * q2,
                             q0 * q0 + q1 * q1 + q2 * q2);
        lg[i]  = gpot[b * NPTS + i] * K_B2;
    }
    for (int i = tid; i < ROWS_PER_BLOCK; i += THREADS) {
        const float p0 = pr[(b * NPTS + n0b + i) * 3 + 0];
        const float p1 = pr[(b * NPTS + n0b + i) * 3 + 1];
        const float p2 = pr[(b * NPTS + n0b + i) * 3 + 2];
        lx0[i] = p0; lx1[i] = p1; lx2[i] = p2;
        lxn[i] = p0 * p0 + p1 * p1 + p2 * p2;
        lf[i]  = fpot[b * NPTS + n0b + i] * K_B2;
    }
    __syncthreads();

    const int lane  = tid & 31;
    const int wave  = tid >> 5;
    const int half  = lane >> 4;
    const int l     = lane & 15;
    const int rbase = wave * 16;

    const float clamp_lo = 1e-12f;

#if HAVE_WMMA_F32X4
    v2f afrag;
    afrag.x = half ? lx2[rbase + l] : lx0[rbase + l];
    afrag.y = half ? 1.0f           : lx1[rbase + l];
    v8f cfrag;
    #pragma unroll
    for (int v = 0; v < 8; ++v) cfrag[v] = lxn[rbase + v + 8 * half];
    const float* bbase = (const float*)lbp + 2 * half;
#else
    float xr0[8], xr1[8], xr2[8], xrn[8];
    #pragma unroll
    for (int v = 0; v < 8; ++v) {
        const int r = rbase + v + 8 * half;
        xr0[v] = lx0[r]; xr1[v] = lx1[r]; xr2[v] = lx2[r]; xrn[v] = lxn[r];
    }
#endif
    float frk[8];
    #pragma unroll
    for (int v = 0; v < 8; ++v) frk[v] = lf[rbase + v + 8 * half];

    float acc = 0.0f;
    #pragma unroll 2
    for (int mt = 0; mt < NPTS / 16; ++mt) {
        const int   m  = mt * 16 + l;
        const float gk = lg[m];
#if HAVE_WMMA_F32X4
        const v2f bfrag = *(const v2f*)(bbase + 4 * m);
        v8f sq = __builtin_amdgcn_wmma_f32_16x16x4_f32(
            false, afrag, false, bfrag, (short)0, cfrag, false, false);
#else
        const float4 yq = lbp[m];
        v8f sq;
        #pragma unroll
        for (int v = 0; v < 8; ++v)
            sq[v] = xrn[v] + yq.w + xr0[v] * yq.x + xr1[v] * yq.y + xr2[v] * yq.z;
#endif
        float d[8];
        #pragma unroll
        for (int v = 0; v < 8; ++v)
            d[v] = fast_sqrt(fmax_raw(sq[v], clamp_lo));
        #pragma unroll
        for (int v = 0; v < 8; ++v) {
            const float t = __builtin_fmaf(d[v], -K_B2, frk[v] + gk);
            const float p = fast_exp2(t);
            acc = __builtin_fmaf(p, d[v], acc);
        }
    }

    #pragma unroll
    for (int off = 16; off >= 1; off >>= 1)
        acc += __shfl_xor(acc, off, 32);
    if (lane == 0) wsum[wave] = acc;
    __syncthreads();
    if (tid == 0) {
        float total = 0.0f;
        #pragma unroll
        for (int w = 0; w < THREADS / 32; ++w) total += wsum[w];
        atomicAdd(out, total * (1.0f / (float)NBATCH));
    }
}

__global__ void init_zero(float* __restrict__ ws, int nws,
                          float* __restrict__ out, int nout)
{
    const int i = blockIdx.x * blockDim.x + threadIdx.x;
    if (i < nws)  ws[i]  = 0.0f;
    if (i < nout) out[i] = 0.0f;
}

extern "C" void kernel_launch(void* const* d_in, const int* in_sizes, int n_in,
                              void* d_out, int out_size, void* d_ws, size_t ws_size,
                              hipStream_t stream)
{
    const float* tpl = (const float*)d_in[0];   // [16, 2048, 3]
    const float* src = (const float*)d_in[1];   // [16, 2048, 3]
    float* out = (float*)d_out;

    float* f = (float*)d_ws;                    // [16, 2048]
    float* g = f + NBATCH * NPTS;               // [16, 2048]
    const int npot = 2 * NBATCH * NPTS;

    init_zero<<<(npot + THREADS - 1) / THREADS, THREADS, 0, stream>>>(f, npot, out, out_size);

    const dim3 grid(NPTS / ROWS_PER_BLOCK, NBATCH);   // (16, 16)
    for (int it = 0; it < 50; ++it) {
        // f-update: rows = template, cols = source, reads g, writes f
        sinkhorn_half<<<grid, THREADS, 0, stream>>>(tpl, src, g, f);
        // g-update: rows = source, cols = template, reads f, writes g
        sinkhorn_half<<<grid, THREADS, 0, stream>>>(src, tpl, f, g);
    }
    emd_loss<<<grid, THREADS, 0, stream>>>(tpl, src, f, g, out);
}